// RotaryPositionalEmbedding_52862457479385
// MI455X (gfx1250) — compile-verified
//
#include <hip/hip_runtime.h>

// RoPE as block-diagonal GEMM on the CDNA5 matrix pipeline.
// out[bh, s, i] = sum_j R[pos(s), i, j] * x[bh, s, j]
// R is block-diagonal 2x2 rotations -> Out_s = X_s * R_s^T decomposes into four
// independent 64x16x16 GEMMs per position; B tiles are synthesized in VGPRs
// from the 32 (cos,sin) pairs so dense R is never read (saves 64MB of traffic).
// X_s is staged into LDS with GLOBAL_LOAD_ASYNC_TO_LDS_B128 (ASYNCcnt path).

typedef __attribute__((ext_vector_type(2))) float v2f;
typedef __attribute__((ext_vector_type(8))) float v8f;

#define DK 64
#define S_LEN 4096            // fixed by the reference; makes store offsets immediates
#define LDS_STRIDE 68         // 64 + 4 pad: keeps 16B alignment, rotates banks by 4/row

__global__ __launch_bounds__(256) void rope_wmma_kernel(
    const float* __restrict__ x,    // [BH=64, S, 64]
    const int*   __restrict__ tpos, // [S]
    const float* __restrict__ R,    // [S, 64, 64]
    float* __restrict__ out)        // [BH=64, S, 64]
{
    __shared__ float tileX[64 * LDS_STRIDE];
    __shared__ float shC[32];
    __shared__ float shS[32];

    const int s = blockIdx.x;
    const int t = threadIdx.x;

    // ---- async, coalesced load of X_s (64 rows x 256B) straight into LDS ----
    // GV mode: imm advances
    // both sides in lockstep, matching our contiguous quarter-row layout.
    {
        const int row = t >> 2;        // 0..63  (flattened b*h)
        const int cq  = (t & 3) << 4;  // 0,16,32,48
        const float* src = x + ((size_t)row * S_LEN + s) * DK + cq;
        // low 32 bits of the flat-cast shared pointer are the LDS byte offset
        uint32_t ldsb = (uint32_t)(uintptr_t)&tileX[row * LDS_STRIDE + cq];
        asm volatile(
            "global_load_async_to_lds_b128 %0, %1, off offset:0\n\t"
            "global_load_async_to_lds_b128 %0, %1, off offset:16\n\t"
            "global_load_async_to_lds_b128 %0, %1, off offset:32\n\t"
            "global_load_async_to_lds_b128 %0, %1, off offset:48"
            :: "v"(ldsb), "v"(src) : "memory");
    }
    // ---- gather the sparse rotation entries: cos = R[p,2p,2p], sin = R[p,2p+1,2p] ----
    if (t < 32) {
        const int pos = tpos[s];
        const float* Rb = R + (size_t)pos * DK * DK;
        shC[t] = Rb[(2 * t) * DK + 2 * t];
        shS[t] = Rb[(2 * t + 1) * DK + 2 * t];
    }
    asm volatile("s_wait_asynccnt 0x0" ::: "memory");
    __syncthreads();

    const int wave = t >> 5;
    const int lane = t & 31;
    const int half = lane >> 4;  // lane half selects K-pairs per ISA f32 layouts
    const int l    = lane & 15;

    // Wave w owns output-column tile i0 and row tiles m0 and m0+32; both row
    // tiles share the same rotation columns -> build B fragments once.
    const int i0 = (wave & 3) * 16;
    const int m0 = (wave >> 2) * 16;

    const float c  = shC[(i0 >> 1) + (l >> 1)];
    const float sv = shS[(i0 >> 1) + (l >> 1)];

    // B (4x16 f32), B = R^T block: B[2p,2p]=c, B[2p,2p+1]=s, B[2p+1,2p]=-s,
    // B[2p+1,2p+1]=c.  VGPR0 row jj0 (even, lanes split by half), VGPR1 row jj0+1.
    v2f bf[4];
    #pragma unroll
    for (int kk = 0; kk < 4; ++kk) {
        const int jj0 = kk * 4 + 2 * half;
        bf[kk].x = (l == jj0)     ? c : ((l == jj0 + 1) ?  sv : 0.0f);
        bf[kk].y = (l == jj0 + 1) ? c : ((l == jj0)     ? -sv : 0.0f);
    }

    v8f acc0 = {};
    v8f acc1 = {};
    #pragma unroll
    for (int kk = 0; kk < 4; ++kk) {
        // A (16x4 f32): lanes0-15 -> {K,K+1}, lanes16-31 -> {K+2,K+3}
        const int colA = i0 + kk * 4 + 2 * half;
        v2f a0 = *(const v2f*)&tileX[(m0      + l) * LDS_STRIDE + colA];
        v2f a1 = *(const v2f*)&tileX[(m0 + 32 + l) * LDS_STRIDE + colA];
        acc0 = __builtin_amdgcn_wmma_f32_16x16x4_f32(
            false, a0, false, bf[kk], (short)0, acc0, false, false);
        acc1 = __builtin_amdgcn_wmma_f32_16x16x4_f32(
            false, a1, false, bf[kk], (short)0, acc1, false, false);
    }

    // D (16x16 f32): VGPR r -> row m0 + r + 8*half, col i0 + l.
    // With S_LEN constexpr, r*S_LEN*DK*4 = r*1MB < 8MB -> immediate offsets.
    float* op0 = out + ((size_t)(m0 + 8 * half) * S_LEN + s) * DK + i0 + l;
    float* op1 = op0 + (size_t)32 * S_LEN * DK;
    #pragma unroll
    for (int r = 0; r < 8; ++r) {
        op0[(size_t)r * S_LEN * DK] = acc0[r];
        op1[(size_t)r * S_LEN * DK] = acc1[r];
    }
}

extern "C" void kernel_launch(void* const* d_in, const int* in_sizes, int n_in,
                              void* d_out, int out_size, void* d_ws, size_t ws_size,
                              hipStream_t stream) {
    const float* x  = (const float*)d_in[0];
    const int*   tp = (const int*)d_in[1];
    const float* R  = (const float*)d_in[2];
    float*       o  = (float*)d_out;
    (void)in_sizes; (void)n_in; (void)out_size; (void)d_ws; (void)ws_size;
    rope_wmma_kernel<<<dim3(S_LEN), dim3(256), 0, stream>>>(x, tp, R, o);
}